// CompressiveMemory_90718299226448
// MI455X (gfx1250) — compile-verified
//
#include <hip/hip_runtime.h>

// ---------------------------------------------------------------------------
// MI455X (gfx1250) CompressiveMemory forward.
//  B=4, T=4096, D=2048, H=16, DK=DV=128, SEG=512, N_SEG=8.
//  f32 end-to-end with V_WMMA_F32_16X16X4_F32.
//  GEMM tiles are streamed global->LDS by the Tensor Data Mover
//  (tensor_load_to_lds descriptors, TENSORcnt-synchronized, double buffered,
//  with DMA-side LDS row padding to the stride-34 fragment layout).
//
//  Convention: feeding both WMMA operands in (rows x contraction) order
//  computes C = X @ Y^T, which this implementation uses to avoid all
//  transposes except a one-time weight transpose and in-LDS V^T staging.
//
//  Workspace (floats):
//    Wt_q @ 0, Wt_k @ 4194304, Wt_v @ 8388608, Wt_o @ 12582912
//    Q @ 16777216, K @ 50331648, V @ 83886080, ATT @ 117440512
//  Total 150,994,944 floats = 603,979,776 bytes required in d_ws.
// ---------------------------------------------------------------------------

typedef float v2f __attribute__((ext_vector_type(2)));
typedef float v8f __attribute__((ext_vector_type(8)));
typedef unsigned int u32x4 __attribute__((ext_vector_type(4)));
typedef unsigned int u32x8 __attribute__((ext_vector_type(8)));

__device__ __forceinline__ v8f vzero8() {
  v8f z;
#pragma unroll
  for (int i = 0; i < 8; ++i) z[i] = 0.f;
  return z;
}

// D = A(16x4) * B(4x16) + C   (f32 WMMA)
__device__ __forceinline__ v8f wmma4(v2f a, v2f b, v8f c) {
  return __builtin_amdgcn_wmma_f32_16x16x4_f32(
      /*neg_a=*/false, a, /*neg_b=*/false, b,
      /*c_mod=*/(short)0, c, /*reuse_a=*/false, /*reuse_b=*/false);
}

// A/B fragment loader from a (rows x contraction) row-major f32 tile.
// lane 0-15: row m0+lane, K = k0,k0+1 ; lane 16-31: row m0+lane-16, K = k0+2,k0+3
__device__ __forceinline__ v2f ldfrag(const float* __restrict__ p, int stride,
                                      int m0, int lane, int k0) {
  return *(const v2f*)(p + (size_t)(m0 + (lane & 15)) * stride +
                       (k0 + ((lane >> 4) << 1)));
}

// ---------------------------------------------------------------------------
// TDM: DMA one 128-row x 32-dword tile (row stride `ld` dwords) from global
// memory into LDS at byte offset `lds_off`, inserting 2 pad dwords after
// every 32 dwords (-> LDS row stride 34 = the WMMA fragment layout).
// D# per CDNA5 ISA, 2-group form (VADDR2/3=NULL).
// Scalar-issued; track with TENSORcnt.
// ---------------------------------------------------------------------------
__device__ __forceinline__ void tdm_tile_128x32(const float* gsrc,
                                                unsigned int lds_off,
                                                unsigned int ld) {
  const unsigned long long ga = (unsigned long long)(uintptr_t)gsrc;
  u32x4 g0;
  g0[0] = 1u;                                   // count=1 user descriptor
  g0[1] = lds_off;                              // lds_addr (bytes)
  g0[2] = (unsigned int)ga;                     // global_addr[31:0]
  g0[3] = (unsigned int)(ga >> 32) | (2u << 30);// global_addr[56:32] | type=2
  u32x8 g1;
  g1[0] = (2u << 16)    // data_size = 4B
        | (1u << 20)    // pad_enable
        | (4u << 22)    // pad_interval: 32 dwords
        | (1u << 25);   // pad_amount:   2 dwords
  g1[1] = (ld & 0xffffu) << 16;                 // tensor_dim0 lo16 (=ld)
  g1[2] = (ld >> 16);                           // tensor_dim0 hi16, dim1 lo16=0
  g1[3] = 16u | (32u << 16);                    // tensor_dim1=1<<20 | tile_dim0=32
  g1[4] = 128u;                                 // tile_dim1=128, tile_dim2=0
  g1[5] = ld;                                   // tensor_dim0_stride lo32
  g1[6] = 0u;                                   // stride hi16 | dim1_stride lo16
  g1[7] = 0u;
  asm volatile("tensor_load_to_lds %0, %1" ::"s"(g0), "s"(g1) : "memory");
}

// ---------------------------------------------------------------------------
// 32x32 tiled transpose: out(C x R) = in(R x C)^T.  R, C multiples of 32.
// ---------------------------------------------------------------------------
__global__ __launch_bounds__(256) void transpose_f32(
    const float* __restrict__ in, float* __restrict__ out, int R, int C) {
  __shared__ float tl[32][33];
  const int tx = threadIdx.x, ty = threadIdx.y;
  const int c0 = blockIdx.x * 32, r0 = blockIdx.y * 32;
#pragma unroll
  for (int i = 0; i < 4; ++i)
    tl[ty + i * 8][tx] = in[(size_t)(r0 + ty + i * 8) * C + c0 + tx];
  __syncthreads();
#pragma unroll
  for (int i = 0; i < 4; ++i)
    out[(size_t)(c0 + ty + i * 8) * R + r0 + tx] = tl[tx][ty + i * 8];
}

// ---------------------------------------------------------------------------
// C(MxN) = A(MxK) @ Bt(NxK)^T, f32 WMMA.  128x128 block tile, BK=32,
// 256 threads = 8 waves, each wave owns 32(M) x 64(N) of accumulators.
// Tiles streamed by TDM, double buffered.  Dynamic LDS: 4 x 4352 floats.
// ---------------------------------------------------------------------------
__global__ __launch_bounds__(256) void gemm_f32_wmma(
    const float* __restrict__ A, const float* __restrict__ Bt,
    float* __restrict__ C, int M, int N, int K) {
  extern __shared__ float gsm[];
  float* As0 = gsm;
  float* As1 = gsm + 4352;
  float* Bs0 = gsm + 2 * 4352;
  float* Bs1 = gsm + 3 * 4352;
  const unsigned int offA0 = (unsigned int)(uintptr_t)As0;
  const unsigned int offA1 = (unsigned int)(uintptr_t)As1;
  const unsigned int offB0 = (unsigned int)(uintptr_t)Bs0;
  const unsigned int offB1 = (unsigned int)(uintptr_t)Bs1;

  const int t = threadIdx.x, lane = t & 31, w = t >> 5;
  const int wm = w >> 1, wn = w & 1;
  const int hs = lane >> 4, n15 = lane & 15;
  const int bm = blockIdx.y * 128, bn = blockIdx.x * 128;

  v8f acc[2][4];
#pragma unroll
  for (int i = 0; i < 2; ++i)
#pragma unroll
    for (int j = 0; j < 4; ++j) acc[i][j] = vzero8();

  // prologue: DMA the first k-slab into buffer 0
  if (w == 0) {
    tdm_tile_128x32(A + (size_t)bm * K, offA0, (unsigned int)K);
    tdm_tile_128x32(Bt + (size_t)bn * K, offB0, (unsigned int)K);
    __builtin_amdgcn_s_wait_tensorcnt(0);
  }
  __syncthreads();

  for (int kb = 0; kb < K; kb += 32) {
    const int cur = (kb >> 5) & 1;
    const float* Ac = cur ? As1 : As0;
    const float* Bc = cur ? Bs1 : Bs0;
    const bool more = (kb + 32 < K);
    if (w == 0 && more) {  // async DMA of the next slab into the other buffer
      tdm_tile_128x32(A + (size_t)bm * K + kb + 32, cur ? offA0 : offA1,
                      (unsigned int)K);
      tdm_tile_128x32(Bt + (size_t)bn * K + kb + 32, cur ? offB0 : offB1,
                      (unsigned int)K);
    }
#pragma unroll
    for (int kk = 0; kk < 32; kk += 4) {
      v2f a0 = ldfrag(Ac, 34, wm * 32, lane, kk);
      v2f a1 = ldfrag(Ac, 34, wm * 32 + 16, lane, kk);
#pragma unroll
      for (int j = 0; j < 4; ++j) {
        v2f bb = ldfrag(Bc, 34, wn * 64 + j * 16, lane, kk);
        acc[0][j] = wmma4(a0, bb, acc[0][j]);
        acc[1][j] = wmma4(a1, bb, acc[1][j]);
      }
    }
    if (w == 0 && more) __builtin_amdgcn_s_wait_tensorcnt(0);
    __syncthreads();  // next buffer landed; this buffer free for reuse
  }
#pragma unroll
  for (int i = 0; i < 2; ++i)
#pragma unroll
    for (int j = 0; j < 4; ++j)
#pragma unroll
      for (int v = 0; v < 8; ++v) {
        const int row = bm + wm * 32 + i * 16 + v + 8 * hs;
        const int col = bn + wn * 64 + j * 16 + n15;
        C[(size_t)row * N + col] = acc[i][j][v];
      }
}

// ---------------------------------------------------------------------------
// Segment-local attention: att_dot = softmax(q k^T / sqrt(DK)) @ v.
// One block = one (b,seg,h) head (contiguous 512x128 blocks) x one 64-row
// q tile.  Exact two-pass online softmax; V^T chunk staged in LDS.
// Dynamic LDS: sc 64x130 | vts 128x130 | rmax 64 | rsum 64 = 100,352 B.
// ---------------------------------------------------------------------------
__global__ __launch_bounds__(256) void attn_dot_kernel(
    const float* __restrict__ Q, const float* __restrict__ Kb,
    const float* __restrict__ Vb, float* __restrict__ Att) {
  extern __shared__ float sm1[];
  float* sc = sm1;                      // 64 x 130 (scores / probs)
  float* vts = sm1 + 64 * 130;          // 128 x 130 (V^T chunk)
  float* rmax = sm1 + 64 * 130 + 128 * 130;
  float* rsum = rmax + 64;

  const int t = threadIdx.x, lane = t & 31, w = t >> 5;
  const int wm = w & 3, wn = w >> 2;  // 16-row m-tile, 64-col n-half
  const int hs = lane >> 4, n15 = lane & 15;
  const int sh = blockIdx.x, rt = blockIdx.y;
  const float* qs = Q + (size_t)sh * 65536;
  const float* ks = Kb + (size_t)sh * 65536;
  const float* vs = Vb + (size_t)sh * 65536;
  float* as = Att + (size_t)sh * 65536;
  const float scale = 0.088388347648318447f;  // 1/sqrt(128)

  if (t < 64) { rmax[t] = -3.0e38f; rsum[t] = 0.f; }
  __syncthreads();

  // ---- phase A: online row max / sum over 4 key chunks of 128
  for (int ch = 0; ch < 4; ++ch) {
    v8f s0 = vzero8(), s1 = vzero8(), s2 = vzero8(), s3 = vzero8();
#pragma unroll 4
    for (int kk = 0; kk < 128; kk += 4) {
      v2f a = ldfrag(qs, 128, rt * 64 + wm * 16, lane, kk);
      s0 = wmma4(a, ldfrag(ks, 128, ch * 128 + wn * 64 + 0, lane, kk), s0);
      s1 = wmma4(a, ldfrag(ks, 128, ch * 128 + wn * 64 + 16, lane, kk), s1);
      s2 = wmma4(a, ldfrag(ks, 128, ch * 128 + wn * 64 + 32, lane, kk), s2);
      s3 = wmma4(a, ldfrag(ks, 128, ch * 128 + wn * 64 + 48, lane, kk), s3);
    }
#pragma unroll
    for (int v = 0; v < 8; ++v) {
      const int row = wm * 16 + v + 8 * hs;
      sc[row * 130 + wn * 64 + 0 + n15] = s0[v];
      sc[row * 130 + wn * 64 + 16 + n15] = s1[v];
      sc[row * 130 + wn * 64 + 32 + n15] = s2[v];
      sc[row * 130 + wn * 64 + 48 + n15] = s3[v];
    }
    __syncthreads();
    if (t < 64) {
      float m = rmax[t], ssum = rsum[t], cm = m;
      for (int c = 0; c < 128; ++c) cm = fmaxf(cm, sc[t * 130 + c] * scale);
      float add = 0.f;
      for (int c = 0; c < 128; ++c) add += __expf(sc[t * 130 + c] * scale - cm);
      rsum[t] = ssum * __expf(m - cm) + add;
      rmax[t] = cm;
    }
    __syncthreads();
  }

  // ---- phase B: P = softmax(scores); out += P @ V (per chunk)
  v8f o0 = vzero8(), o1 = vzero8(), o2 = vzero8(), o3 = vzero8();
  for (int ch = 0; ch < 4; ++ch) {
#pragma unroll
    for (int i = 0; i < 64; ++i) {  // stage V^T chunk (scatter into LDS)
      const int idx = t + 256 * i;
      const int key = idx >> 7, dv = idx & 127;
      vts[dv * 130 + key] = vs[(size_t)(ch * 128 + key) * 128 + dv];
    }
    v8f s0 = vzero8(), s1 = vzero8(), s2 = vzero8(), s3 = vzero8();
#pragma unroll 4
    for (int kk = 0; kk < 128; kk += 4) {
      v2f a = ldfrag(qs, 128, rt * 64 + wm * 16, lane, kk);
      s0 = wmma4(a, ldfrag(ks, 128, ch * 128 + wn * 64 + 0, lane, kk), s0);
      s1 = wmma4(a, ldfrag(ks, 128, ch * 128 + wn * 64 + 16, lane, kk), s1);
      s2 = wmma4(a, ldfrag(ks, 128, ch * 128 + wn * 64 + 32, lane, kk), s2);
      s3 = wmma4(a, ldfrag(ks, 128, ch * 128 + wn * 64 + 48, lane, kk), s3);
    }
#pragma unroll
    for (int v = 0; v < 8; ++v) {
      const int row = wm * 16 + v + 8 * hs;
      const float mx = rmax[row], inv = 1.f / rsum[row];
      sc[row * 130 + wn * 64 + 0 + n15] = __expf(s0[v] * scale - mx) * inv;
      sc[row * 130 + wn * 64 + 16 + n15] = __expf(s1[v] * scale - mx) * inv;
      sc[row * 130 + wn * 64 + 32 + n15] = __expf(s2[v] * scale - mx) * inv;
      sc[row * 130 + wn * 64 + 48 + n15] = __expf(s3[v] * scale - mx) * inv;
    }
    __syncthreads();
#pragma unroll 4
    for (int kk = 0; kk < 128; kk += 4) {  // out += P @ (V^T)^T
      v2f a = ldfrag(sc, 130, wm * 16, lane, kk);
      o0 = wmma4(a, ldfrag(vts, 130, wn * 64 + 0, lane, kk), o0);
      o1 = wmma4(a, ldfrag(vts, 130, wn * 64 + 16, lane, kk), o1);
      o2 = wmma4(a, ldfrag(vts, 130, wn * 64 + 32, lane, kk), o2);
      o3 = wmma4(a, ldfrag(vts, 130, wn * 64 + 48, lane, kk), o3);
    }
    __syncthreads();
  }
#pragma unroll
  for (int v = 0; v < 8; ++v) {
    const int row = rt * 64 + wm * 16 + v + 8 * hs;
    as[(size_t)row * 128 + wn * 64 + 0 + n15] = o0[v];
    as[(size_t)row * 128 + wn * 64 + 16 + n15] = o1[v];
    as[(size_t)row * 128 + wn * 64 + 32 + n15] = o2[v];
    as[(size_t)row * 128 + wn * 64 + 48 + n15] = o3[v];
  }
}

// ---------------------------------------------------------------------------
// Recurrent compressive-memory scan. One block per (b,h); sequential over 8
// segments.  State memT (DV x DK) lives in WMMA accumulator fragments
// (registers) the whole kernel; published to LDS once per segment for use
// as a GEMM operand. z collapses to a 128-vector (broadcast column sums).
// Dynamic LDS: memT | sq | sq^T | sk/U^T (each 128x130) + 5x128 = 268,800 B.
// ---------------------------------------------------------------------------
__global__ __launch_bounds__(256) void recur_kernel(
    const float* __restrict__ Q, const float* __restrict__ Kb,
    const float* __restrict__ Vb, const float* __restrict__ Betas,
    float* __restrict__ Att) {
  extern __shared__ float sm2[];
  float* memT = sm2;                 // 128 x 130  (dv, dk)
  float* bufA = sm2 + 16640;         // sq  (s, dk)
  float* bufAT = sm2 + 2 * 16640;    // sq^T (dk, s)
  float* bufB = sm2 + 3 * 16640;     // sk (s,dk) then U^T (dv, s)
  float* zv = sm2 + 4 * 16640;       // 128
  float* beta = zv + 128;            // 128
  float* rsq = zv + 256;             // 128
  float* rsk = zv + 384;             // 128
  float* kcol = zv + 512;            // 128

  const int t = threadIdx.x, lane = t & 31, w = t >> 5;
  const int wm = w >> 1, wn = w & 1;
  const int hs = lane >> 4, n15 = lane & 15;
  const int b = blockIdx.x >> 4, h = blockIdx.x & 15;

  v8f memf[2][4];
#pragma unroll
  for (int i = 0; i < 2; ++i)
#pragma unroll
    for (int j = 0; j < 4; ++j) memf[i][j] = vzero8();

  if (t < 128) {
    zv[t] = 1.f;
    beta[t] = 1.f / (1.f + __expf(-Betas[h * 128 + t]));
  }
  __syncthreads();

  for (int n = 0; n < 8; ++n) {
    const size_t base = (size_t)(b * 128 + n * 16 + h) * 65536;
    const float* qs = Q + base;
    const float* ks = Kb + base;
    const float* vs = Vb + base;
    float* as = Att + base;

    // publish OLD state to LDS for this segment
#pragma unroll
    for (int i = 0; i < 2; ++i)
#pragma unroll
      for (int j = 0; j < 4; ++j)
#pragma unroll
        for (int v = 0; v < 8; ++v)
          memT[(wm * 32 + i * 16 + v + 8 * hs) * 130 + wn * 64 + j * 16 + n15] =
              memf[i][j][v];
    if (t < 128) kcol[t] = 0.f;
    __syncthreads();

    for (int ch = 0; ch < 4; ++ch) {
      // sk -> bufB ; sq -> bufA (natural) + bufAT (transposed)
#pragma unroll
      for (int i = 0; i < 64; ++i) {
        const int idx = t + 256 * i;
        const int s = idx >> 7, dk = idx & 127;
        const float xk = ks[(size_t)(ch * 128 + s) * 128 + dk];
        bufB[s * 130 + dk] = (xk > 0.f) ? xk + 1.f : __expf(xk);
        const float xq = qs[(size_t)(ch * 128 + s) * 128 + dk];
        const float sv = (xq > 0.f) ? xq + 1.f : __expf(xq);
        bufA[s * 130 + dk] = sv;
        bufAT[dk * 130 + s] = sv;
      }
      __syncthreads();
      if (t < 128) {
        float a = 0.f, bs = 0.f, c = 0.f;
        for (int q2 = 0; q2 < 128; ++q2) {
          a += bufB[t * 130 + q2];   // row sums of sk   (per s)
          bs += bufB[q2 * 130 + t];  // col sums of sk   (per dk, -> z)
          c += bufA[t * 130 + q2];   // row sums of sq   (per s)
        }
        rsk[t] = a;
        kcol[t] += bs;
        rsq[t] = c;
      }
      __syncthreads();

      // corr = sk @ mem = bufB @ memT^T
      v8f cf[2][4];
#pragma unroll
      for (int i = 0; i < 2; ++i)
#pragma unroll
        for (int j = 0; j < 4; ++j) cf[i][j] = vzero8();
#pragma unroll 4
      for (int kk = 0; kk < 128; kk += 4) {
        v2f a0 = ldfrag(bufB, 130, wm * 32, lane, kk);
        v2f a1 = ldfrag(bufB, 130, wm * 32 + 16, lane, kk);
#pragma unroll
        for (int j = 0; j < 4; ++j) {
          v2f bb = ldfrag(memT, 130, wn * 64 + j * 16, lane, kk);
          cf[0][j] = wmma4(a0, bb, cf[0][j]);
          cf[1][j] = wmma4(a1, bb, cf[1][j]);
        }
      }
      __syncthreads();  // bufB (sk) now free for reuse as U^T

      // U = v - corr / (rowsum_sk * z)  -> stored transposed in bufB
#pragma unroll
      for (int i = 0; i < 2; ++i)
#pragma unroll
        for (int j = 0; j < 4; ++j)
#pragma unroll
          for (int v = 0; v < 8; ++v) {
            const int s = wm * 32 + i * 16 + v + 8 * hs;
            const int dv = wn * 64 + j * 16 + n15;
            const float u = vs[(size_t)(ch * 128 + s) * 128 + dv] -
                            cf[i][j][v] / (rsk[s] * zv[dv]);
            bufB[dv * 130 + s] = u;
          }

      // att_mem numerator = sq @ mem = bufA @ memT^T (reuse cf regs)
#pragma unroll
      for (int i = 0; i < 2; ++i)
#pragma unroll
        for (int j = 0; j < 4; ++j) cf[i][j] = vzero8();
#pragma unroll 4
      for (int kk = 0; kk < 128; kk += 4) {
        v2f a0 = ldfrag(bufA, 130, wm * 32, lane, kk);
        v2f a1 = ldfrag(bufA, 130, wm * 32 + 16, lane, kk);
#pragma unroll
        for (int j = 0; j < 4; ++j) {
          v2f bb = ldfrag(memT, 130, wn * 64 + j * 16, lane, kk);
          cf[0][j] = wmma4(a0, bb, cf[0][j]);
          cf[1][j] = wmma4(a1, bb, cf[1][j]);
        }
      }
      // blend with att_dot in place
#pragma unroll
      for (int i = 0; i < 2; ++i)
#pragma unroll
        for (int j = 0; j < 4; ++j)
#pragma unroll
          for (int v = 0; v < 8; ++v) {
            const int s = wm * 32 + i * 16 + v + 8 * hs;
            const int dv = wn * 64 + j * 16 + n15;
            const float am = cf[i][j][v] / (rsq[s] * zv[dv]);
            const size_t gi = (size_t)(ch * 128 + s) * 128 + dv;
            as[gi] = beta[dv] * am + (1.f - beta[dv]) * as[gi];
          }
      __syncthreads();  // all U^T writes visible

      // memT_frags += U^T @ (sq^T)^T   (delta rule, old mem used above)
#pragma unroll 4
      for (int kk = 0; kk < 128; kk += 4) {
        v2f a0 = ldfrag(bufB, 130, wm * 32, lane, kk);
        v2f a1 = ldfrag(bufB, 130, wm * 32 + 16, lane, kk);
#pragma unroll
        for (int j = 0; j < 4; ++j) {
          v2f bb = ldfrag(bufAT, 130, wn * 64 + j * 16, lane, kk);
          memf[0][j] = wmma4(a0, bb, memf[0][j]);
          memf[1][j] = wmma4(a1, bb, memf[1][j]);
        }
      }
      __syncthreads();  // buffers free for next chunk
    }
    if (t < 128) zv[t] += kcol[t];  // z update with this segment's sk sums
    __syncthreads();
  }
}

// ---------------------------------------------------------------------------
extern "C" void kernel_launch(void* const* d_in, const int* in_sizes, int n_in,
                              void* d_out, int out_size, void* d_ws,
                              size_t ws_size, hipStream_t stream) {
  (void)in_sizes; (void)n_in; (void)out_size; (void)ws_size;
  const float* x = (const float*)d_in[0];
  const float* Wq = (const float*)d_in[1];
  const float* Wk = (const float*)d_in[2];
  const float* Wv = (const float*)d_in[3];
  const float* Wo = (const float*)d_in[4];
  const float* betas = (const float*)d_in[5];
  float* out = (float*)d_out;

  float* ws = (float*)d_ws;
  float* Wtq = ws;
  float* Wtk = ws + 4194304;
  float* Wtv = ws + 8388608;
  float* Wto = ws + 12582912;
  float* Qb = ws + 16777216;
  float* Kbf = ws + 16777216 + 33554432;
  float* Vbf = ws + 16777216 + 2 * 33554432;
  float* Atb = ws + 16777216 + 3 * 33554432;

  // 1) transpose the four weight matrices (one-time, 16 MB each)
  dim3 tb(32, 8), tg(64, 64);
  transpose_f32<<<tg, tb, 0, stream>>>(Wq, Wtq, 2048, 2048);
  transpose_f32<<<tg, tb, 0, stream>>>(Wk, Wtk, 2048, 2048);
  transpose_f32<<<tg, tb, 0, stream>>>(Wv, Wtv, 2048, 2048);
  transpose_f32<<<tg, tb, 0, stream>>>(Wo, Wto, 2048, 2048);

  // 2) q/k/v projections (TDM-fed WMMA GEMMs): (16384x2048)@(2048x2048)
  const size_t gemm_lds = (size_t)4 * 4352 * 4;  // 69,632 B (double buffered)
  hipFuncSetAttribute((const void*)gemm_f32_wmma,
                      hipFuncAttributeMaxDynamicSharedMemorySize,
                      (int)gemm_lds);
  dim3 gg(16, 128);
  gemm_f32_wmma<<<gg, 256, gemm_lds, stream>>>(x, Wtq, Qb, 16384, 2048, 2048);
  gemm_f32_wmma<<<gg, 256, gemm_lds, stream>>>(x, Wtk, Kbf, 16384, 2048, 2048);
  gemm_f32_wmma<<<gg, 256, gemm_lds, stream>>>(x, Wtv, Vbf, 16384, 2048, 2048);

  // 3) segment-local softmax attention (parallel over all 512 seg-heads)
  const size_t s1_lds = (size_t)(64 * 130 + 128 * 130 + 64 + 64) * 4;  // 100352
  hipFuncSetAttribute((const void*)attn_dot_kernel,
                      hipFuncAttributeMaxDynamicSharedMemorySize, (int)s1_lds);
  attn_dot_kernel<<<dim3(512, 8), 256, s1_lds, stream>>>(Qb, Kbf, Vbf, Atb);

  // 4) compressive-memory recurrence + blend (64 blocks, one per (b,h))
  const size_t s2_lds = (size_t)(4 * 16640 + 5 * 128) * 4;  // 268800
  hipFuncSetAttribute((const void*)recur_kernel,
                      hipFuncAttributeMaxDynamicSharedMemorySize, (int)s2_lds);
  recur_kernel<<<dim3(64), 256, s2_lds, stream>>>(Qb, Kbf, Vbf, betas, Atb);

  // 5) output projection: att @ Wo
  gemm_f32_wmma<<<gg, 256, gemm_lds, stream>>>(Atb, Wto, out, 16384, 2048, 2048);
}